// PoissonEMModule_87170656240432
// MI455X (gfx1250) — compile-verified
//
#include <hip/hip_runtime.h>

#define N1    200
#define N2    200
#define NPIX  (N1*N2)        // 40000 per image
#define NRAD  223
#define NPHI  190
#define NSINO (NPHI*NRAD)    // 42370 per batch
#define NT    221
#define NBATCH 2
#define STEPW 2.0f           // 2*220/(221-1)
#define PHI0  0.0165346982f  // pi/190
#define DR    (400.0f/222.0f)

#define PSTRIDE 204          // padded LDS row stride in floats (16B-aligned rows)
#define PROJ_WAVES 16
#define PROJ_BLOCK (PROJ_WAVES*32)
#define GROUPS_PER_PHI 14            // ceil(223/16)
#define TASKS_PER_B (NPHI*GROUPS_PER_PHI)               // 2660
#define PROJ_BLKS_PER_B ((TASKS_PER_B + PROJ_WAVES - 1)/PROJ_WAVES) // 167
#define BP_BLKS_PER_B 32

typedef __attribute__((ext_vector_type(2))) float v2f;
typedef __attribute__((ext_vector_type(8))) float v8f;

__device__ __forceinline__ int clampi(int v, int lo, int hi) {
    return min(max(v, lo), hi);
}

__device__ __forceinline__ void gaussk(float* k) {
    // 1/SIGMA_VOX = 2.35*VOX/4.5 = 4.7/4.5
    const float inv_sigma = 4.7f / 4.5f;
    float s = 0.0f;
#pragma unroll
    for (int d = -4; d <= 4; ++d) {
        float z = (float)d * inv_sigma;
        float v = __expf(-0.5f * z * z);
        k[d + 4] = v; s += v;
    }
    float inv = 1.0f / s;
#pragma unroll
    for (int i = 0; i < 9; ++i) k[i] *= inv;
}

// ---------------- separable Gaussian smoothing (branchless taps) ----------------
__global__ __launch_bounds__(256) void smooth_v_kernel(const float* __restrict__ in,
                                                       float* __restrict__ out) {
    int n = blockIdx.x * 256 + threadIdx.x;
    if (n >= NBATCH * NPIX) return;
    int b = n / NPIX, ij = n - b * NPIX;
    int i = ij / N2, j = ij - i * N2;
    float k[9]; gaussk(k);
    const float* base = in + b * NPIX;
    float a = 0.0f;
#pragma unroll
    for (int d = -4; d <= 4; ++d) {
        int ii = i + d;
        float w = ((unsigned)ii < (unsigned)N1) ? k[d + 4] : 0.0f;
        a += w * base[clampi(ii, 0, N1 - 1) * N2 + j];
    }
    out[n] = a;
}

__global__ __launch_bounds__(256) void smooth_h_kernel(const float* __restrict__ in,
                                                       float* __restrict__ out) {
    int n = blockIdx.x * 256 + threadIdx.x;
    if (n >= NBATCH * NPIX) return;
    int b = n / NPIX, ij = n - b * NPIX;
    int i = ij / N2, j = ij - i * N2;
    float k[9]; gaussk(k);
    const float* base = in + b * NPIX + i * N2;
    float a = 0.0f;
#pragma unroll
    for (int d = -4; d <= 4; ++d) {
        int jj = j + d;
        float w = ((unsigned)jj < (unsigned)N2) ? k[d + 4] : 0.0f;
        a += w * base[clampi(jj, 0, N2 - 1)];
    }
    out[n] = a;
}

// last smoothing pass fused with  out = x * (AT(ratio) / adjoint_ones)
__global__ __launch_bounds__(256) void smooth_h_final_kernel(const float* __restrict__ tmp,
                                                             const float* __restrict__ x,
                                                             const float* __restrict__ aj,
                                                             float* __restrict__ out) {
    int n = blockIdx.x * 256 + threadIdx.x;
    if (n >= NBATCH * NPIX) return;
    int b = n / NPIX, ij = n - b * NPIX;
    int i = ij / N2, j = ij - i * N2;
    float k[9]; gaussk(k);
    const float* base = tmp + b * NPIX + i * N2;
    float a = 0.0f;
#pragma unroll
    for (int d = -4; d <= 4; ++d) {
        int jj = j + d;
        float w = ((unsigned)jj < (unsigned)N2) ? k[d + 4] : 0.0f;
        a += w * base[clampi(jj, 0, N2 - 1)];
    }
    out[n] = x[n] * (a / aj[n]);
}

// ---------------- forward projection + ratio (LDS image, async stage, WMMA t-sum) ----------------
__global__ __launch_bounds__(PROJ_BLOCK) void project_ratio_kernel(
        const float* __restrict__ S, const float* __restrict__ data,
        const float* __restrict__ mult, const float* __restrict__ add,
        float* __restrict__ ratio)
{
    __shared__ float simg[N1 * PSTRIDE];   // 163,200 B < 320 KB WGP LDS

    const int b = blockIdx.x / PROJ_BLKS_PER_B;
    const float* Sb = S + b * NPIX;

    // Async DMA the whole 160 KB image into LDS: one b128 per row-aligned float4.
    for (int c = threadIdx.x; c < NPIX / 4; c += PROJ_BLOCK) {
        int row  = c / (N2 / 4);
        int col4 = (c - row * (N2 / 4)) * 4;
        unsigned long long ga = (unsigned long long)(Sb + row * N2 + col4);
        unsigned la = (unsigned)(size_t)(&simg[row * PSTRIDE + col4]); // low 32b of flat addr == LDS offset
        asm volatile("global_load_async_to_lds_b128 %0, %1, off"
                     :: "v"(la), "v"(ga) : "memory");
    }
    asm volatile("s_wait_asynccnt 0" ::: "memory");
    __syncthreads();

    const int wave = threadIdx.x >> 5;
    const int lane = threadIdx.x & 31;
    const int task = (blockIdx.x % PROJ_BLKS_PER_B) * PROJ_WAVES + wave; // wave-uniform
    if (task >= TASKS_PER_B) return;

    const int p   = task / GROUPS_PER_PHI;
    const int g14 = task - p * GROUPS_PER_PHI;
    float sn, cn;
    sincosf((float)p * PHI0, &sn, &cn);

    // A-matrix 16x4 f32 layout: lanes 0-15 -> M=lane, K={0,1}; lanes 16-31 -> M=lane-16, K={2,3}
    const int m   = lane & 15;
    const int rr  = g14 * 16 + m;
    const float r = -200.0f + (float)(rr < NRAD ? rr : NRAD - 1) * DR;
    const int klo = (lane >> 4) << 1;

    v8f acc = {0.f, 0.f, 0.f, 0.f, 0.f, 0.f, 0.f, 0.f};
    const v2f ones = {1.0f, 1.0f};   // B = all-ones => D rows hold exact f32 row sums

    for (int ch = 0; ch < (NT + 3) / 4; ++ch) {   // 56 chunks of K=4
        v2f a;
#pragma unroll
        for (int u = 0; u < 2; ++u) {
            int tt = ch * 4 + klo + u;
            float tvalid = (tt < NT) ? 1.0f : 0.0f;
            float t  = -220.0f + (float)tt * 2.0f;
            float px = -r * sn + t * cn;
            float py =  r * cn + t * sn;
            float fi = (px + 199.0f) * 0.5f;
            float fj = (py + 199.0f) * 0.5f;
            int i0 = (int)floorf(fi);
            int j0 = (int)floorf(fj);
            float di = fi - (float)i0;
            float dj = fj - (float)j0;
            int i1 = i0 + 1, j1 = j0 + 1;
            bool iA = (unsigned)i0 < (unsigned)N1, iB = (unsigned)i1 < (unsigned)N1;
            bool jA = (unsigned)j0 < (unsigned)N2, jB = (unsigned)j1 < (unsigned)N2;
            // branchless: clamp addresses, zero the weights instead of the loads
            int i0c = clampi(i0, 0, N1 - 1) * PSTRIDE;
            int i1c = clampi(i1, 0, N1 - 1) * PSTRIDE;
            int j0c = clampi(j0, 0, N2 - 1);
            int j1c = clampi(j1, 0, N2 - 1);
            float w00 = (iA && jA) ? (1.0f - di) * (1.0f - dj) : 0.0f;
            float w10 = (iB && jA) ? di * (1.0f - dj)          : 0.0f;
            float w01 = (iA && jB) ? (1.0f - di) * dj          : 0.0f;
            float w11 = (iB && jB) ? di * dj                   : 0.0f;
            float val = tvalid * (w00 * simg[i0c + j0c] + w10 * simg[i1c + j0c]
                                + w01 * simg[i0c + j1c] + w11 * simg[i1c + j1c]);
            if (u == 0) a.x = val; else a.y = val;
        }
        // wave-uniform point, EXEC all-ones here
        acc = __builtin_amdgcn_wmma_f32_16x16x4_f32(false, a, false, ones,
                                                    (short)0, acc, false, false);
    }

    // C/D layout: VGPR v holds row v (lanes 0-15) and row v+8 (lanes 16-31)
#pragma unroll
    for (int v = 0; v < 8; ++v) {
        if ((lane & 15) == v) {
            int ray = v + ((lane >> 4) << 3);
            int rro = g14 * 16 + ray;
            if (rro < NRAD) {
                float proj = acc[v] * STEPW;
                int si = b * NSINO + p * NRAD + rro;
                float mu = mult[si];
                ratio[si] = mu * data[si] / (mu * proj + add[si]);
            }
        }
    }
}

// ---------------- adjoint: branchless bilinear scatter into LDS image, then merge ----------------
__global__ __launch_bounds__(256) void backproject_kernel(const float* __restrict__ ratio,
                                                          float* __restrict__ bp) {
    __shared__ float acc[NPIX];    // 160,000 B
    for (int i = threadIdx.x; i < NPIX; i += 256) acc[i] = 0.0f;
    __syncthreads();

    const int b   = blockIdx.x / BP_BLKS_PER_B;
    const int blk = blockIdx.x % BP_BLKS_PER_B;

    for (int q = blk * 256 + threadIdx.x; q < NSINO; q += BP_BLKS_PER_B * 256) {
        int p  = q / NRAD;
        int rr = q - p * NRAD;
        float rv = ratio[b * NSINO + q] * STEPW;
        float sn, cn;
        sincosf((float)p * PHI0, &sn, &cn);
        float r = -200.0f + (float)rr * DR;
        for (int tt = 0; tt < NT; ++tt) {
            float t  = -220.0f + (float)tt * 2.0f;
            float px = -r * sn + t * cn;
            float py =  r * cn + t * sn;
            float fi = (px + 199.0f) * 0.5f;
            float fj = (py + 199.0f) * 0.5f;
            int i0 = (int)floorf(fi), j0 = (int)floorf(fj);
            float di = fi - (float)i0, dj = fj - (float)j0;
            int i1 = i0 + 1, j1 = j0 + 1;
            bool iA = (unsigned)i0 < (unsigned)N1, iB = (unsigned)i1 < (unsigned)N1;
            bool jA = (unsigned)j0 < (unsigned)N2, jB = (unsigned)j1 < (unsigned)N2;
            int i0c = clampi(i0, 0, N1 - 1) * N2;
            int i1c = clampi(i1, 0, N1 - 1) * N2;
            int j0c = clampi(j0, 0, N2 - 1);
            int j1c = clampi(j1, 0, N2 - 1);
            float w00 = (iA && jA) ? rv * (1.0f - di) * (1.0f - dj) : 0.0f;
            float w10 = (iB && jA) ? rv * di * (1.0f - dj)          : 0.0f;
            float w01 = (iA && jB) ? rv * (1.0f - di) * dj          : 0.0f;
            float w11 = (iB && jB) ? rv * di * dj                   : 0.0f;
            atomicAdd(&acc[i0c + j0c], w00);
            atomicAdd(&acc[i1c + j0c], w10);
            atomicAdd(&acc[i0c + j1c], w01);
            atomicAdd(&acc[i1c + j1c], w11);
        }
    }
    __syncthreads();
    for (int i = threadIdx.x; i < NPIX; i += 256) {
        float v = acc[i];
        if (v != 0.0f) atomicAdd(&bp[b * NPIX + i], v);
    }
}

extern "C" void kernel_launch(void* const* d_in, const int* in_sizes, int n_in,
                              void* d_out, int out_size, void* d_ws, size_t ws_size,
                              hipStream_t stream) {
    const float* x    = (const float*)d_in[0];   // (2,1,200,200)
    const float* data = (const float*)d_in[1];   // (2,190,223)
    const float* mult = (const float*)d_in[2];   // (2,190,223)
    const float* add  = (const float*)d_in[3];   // (2,190,223)
    const float* aj   = (const float*)d_in[4];   // (2,1,200,200)
    float* out = (float*)d_out;                  // (2,1,200,200)

    float* ws = (float*)d_ws;
    float* S   = ws;                 // 80000: smoothed image
    float* T   = ws + 80000;         // 80000: conv temp
    float* R   = ws + 160000;        // 84740: ratio sinogram
    float* BP  = ws + 245000;        // 80000: backprojection accumulator

    const int SM_GRID = (NBATCH * NPIX + 255) / 256;   // 313

    smooth_v_kernel<<<SM_GRID, 256, 0, stream>>>(x, T);
    smooth_h_kernel<<<SM_GRID, 256, 0, stream>>>(T, S);
    hipMemsetAsync(BP, 0, (size_t)NBATCH * NPIX * sizeof(float), stream);
    project_ratio_kernel<<<NBATCH * PROJ_BLKS_PER_B, PROJ_BLOCK, 0, stream>>>(S, data, mult, add, R);
    backproject_kernel<<<NBATCH * BP_BLKS_PER_B, 256, 0, stream>>>(R, BP);
    smooth_v_kernel<<<SM_GRID, 256, 0, stream>>>(BP, T);
    smooth_h_final_kernel<<<SM_GRID, 256, 0, stream>>>(T, x, aj, out);
}